// RNN_83210696392722
// MI455X (gfx1250) — compile-verified
//
#include <hip/hip_runtime.h>
#include <hip/hip_bf16.h>

// ---------------------------------------------------------------------------
// GRU on MI455X (gfx1250, wave32, WMMA + Tensor Data Mover).
//   Kernel A: x_proj = x @ W_x + b   — fp32 WMMA GEMM, TDM-staged, double-buffered
//   Kernel B: persistent cooperative recurrence, W_h resident in LDS (TDM
//             preload), per-step h reload via TDM, one grid sync per timestep.
// ---------------------------------------------------------------------------

typedef float v2f __attribute__((ext_vector_type(2)));
typedef float v8f __attribute__((ext_vector_type(8)));
typedef unsigned int v4u __attribute__((ext_vector_type(4)));
typedef int v4i __attribute__((ext_vector_type(4)));
typedef int v8i __attribute__((ext_vector_type(8)));

#define B_    32
#define T_    2048
#define D_    256
#define H_    256
#define G_    768          // 3*H
#define M_    (B_ * T_)    // 65536

// ------------------------- TDM descriptor helper ---------------------------
// Builds a 2D-tile D# (ISA cdna5 §8) and issues TENSOR_LOAD_TO_LDS.
//   lds_off   : destination byte offset within the workgroup's LDS
//   gptr      : global address of tile start (bytes)
//   tile_x/y  : tile dims in elements (x contiguous)
//   stride_x  : tensor_dim0_stride in elements
//   pad_*     : LDS padding (interval code: 0=2..7=256 DWORDs; amount code: n -> n+1 DWORDs)
__device__ __forceinline__ void tdm_load_2d(unsigned lds_off, const void* gptr,
                                            unsigned tile_x, unsigned tile_y,
                                            unsigned stride_x,
                                            unsigned pad_en, unsigned pad_int_code,
                                            unsigned pad_amt_code)
{
    const unsigned long long ga = (unsigned long long)(size_t)gptr;
    const unsigned tdim = 0x100000u;          // generous dims: no OOB zero-fill
    v4u g0;
    g0[0] = 1u;                                              // count=1, user mode
    g0[1] = lds_off;                                         // lds_addr (bytes)
    g0[2] = (unsigned)(ga & 0xFFFFFFFFu);                    // global_addr lo
    g0[3] = (unsigned)((ga >> 32) & 0x01FFFFFFu) | (2u << 30); // addr hi | type=2
    v8i g1;
    g1[0] = (int)((2u << 16) |                               // data_size = 4B
                  (pad_en << 20) | (pad_int_code << 22) | (pad_amt_code << 25));
    g1[1] = (int)((tdim & 0xFFFFu) << 16);                   // tensor_dim0[15:0]
    g1[2] = (int)(((tdim >> 16) & 0xFFFFu) |                 // tensor_dim0[31:16]
                  ((tdim & 0xFFFFu) << 16));                 // tensor_dim1[15:0]
    g1[3] = (int)(((tdim >> 16) & 0xFFFFu) |                 // tensor_dim1[31:16]
                  ((tile_x & 0xFFFFu) << 16));               // tile_dim0
    g1[4] = (int)(tile_y & 0xFFFFu);                         // tile_dim1 (tile_dim2=0)
    g1[5] = (int)stride_x;                                   // tensor_dim0_stride lo
    g1[6] = 0;                                               // stride hi / dim1_stride
    g1[7] = 0;
    const v4i gz = {0, 0, 0, 0};
#if defined(__clang_major__) && (__clang_major__ >= 23)
    const v8i gz8 = {0, 0, 0, 0, 0, 0, 0, 0};
    __builtin_amdgcn_tensor_load_to_lds(g0, g1, gz, gz, gz8, 0);
#else
    __builtin_amdgcn_tensor_load_to_lds(g0, g1, gz, gz, 0);
#endif
}

// ------------------------- Kernel A: x_proj GEMM ---------------------------
// Tiles: 64 (M) x 128 (N), K-chunks of 32, double-buffered via TDM.
// 8 waves in 2x4 grid, each wave a 32x32 patch (4 fp32 WMMA accumulators).
#define GA_MT 64
#define GA_NT 128
#define GA_KC 32
#define LDA   36            // padded LDS stride for A (bank-conflict-free frags)
#define LDB   136           // padded LDS stride for B
// dynamic LDS float offsets
#define AS0_F 0
#define AS1_F (GA_MT * LDA)                 // 2304
#define BS0_F (2 * GA_MT * LDA)            // 4608
#define BS1_F (2 * GA_MT * LDA + GA_KC * LDB)  // 8960
#define GA_LDS_FLOATS (2 * GA_MT * LDA + 2 * GA_KC * LDB)  // 13312 -> 53,248 B

__global__ __launch_bounds__(256)
void xproj_gemm(const float* __restrict__ x, const float* __restrict__ Wx,
                const float* __restrict__ bias, float* __restrict__ xp)
{
    extern __shared__ float sm[];

    const int tid  = threadIdx.x;
    const int wave = tid >> 5;
    const int lane = tid & 31;
    const int l    = lane & 15;
    const int hi   = lane >> 4;
    const int wm   = wave >> 2;   // 0..1  (M)
    const int wn   = wave & 3;    // 0..3  (N)

    const int m0 = blockIdx.y * GA_MT;
    const int n0 = blockIdx.x * GA_NT;

    const int wave_s = __builtin_amdgcn_readfirstlane(wave);  // scalar branch guard

    // prologue: TDM-stage chunk 0 into buffer 0 (A: 64x32 pad 32->36; B: 32x128 pad 128->136)
    if (wave_s == 0) {
        tdm_load_2d(AS0_F * 4, x + (size_t)m0 * D_, GA_KC, GA_MT, D_, 1, 4, 3);
        tdm_load_2d(BS0_F * 4, Wx + n0,            GA_NT, GA_KC, G_, 1, 6, 7);
    }

    v8f acc[2][2] = {};

    #pragma unroll 1
    for (int c = 0; c < D_ / GA_KC; ++c) {
        if (wave_s == 0) {
            if (c + 1 < D_ / GA_KC) {
                // issue next chunk into the other buffer (free since barrier of c-1)
                const int kc = (c + 1) * GA_KC;
                const unsigned aF = ((c + 1) & 1) ? AS1_F : AS0_F;
                const unsigned bF = ((c + 1) & 1) ? BS1_F : BS0_F;
                tdm_load_2d(aF * 4, x + (size_t)m0 * D_ + kc,  GA_KC, GA_MT, D_, 1, 4, 3);
                tdm_load_2d(bF * 4, Wx + (size_t)kc * G_ + n0, GA_NT, GA_KC, G_, 1, 6, 7);
                __builtin_amdgcn_s_wait_tensorcnt(2);  // chunk c landed (in-order)
            } else {
                __builtin_amdgcn_s_wait_tensorcnt(0);
            }
        }
        __syncthreads();

        const float* As = sm + ((c & 1) ? AS1_F : AS0_F);
        const float* Bs = sm + ((c & 1) ? BS1_F : BS0_F);

        #pragma unroll
        for (int kk = 0; kk < GA_KC; kk += 4) {
            v2f aFr[2], bFr[2];
            #pragma unroll
            for (int ms = 0; ms < 2; ++ms)   // lanes 0-15: K=kk..kk+1; 16-31: K=kk+2..kk+3
                aFr[ms] = *(const v2f*)(As + (wm * 32 + ms * 16 + l) * LDA + kk + 2 * hi);
            #pragma unroll
            for (int ns = 0; ns < 2; ++ns) {
                const int col = wn * 32 + ns * 16 + l;
                bFr[ns].x = Bs[(kk + 2 * hi)     * LDB + col];
                bFr[ns].y = Bs[(kk + 2 * hi + 1) * LDB + col];
            }
            #pragma unroll
            for (int ms = 0; ms < 2; ++ms)
                #pragma unroll
                for (int ns = 0; ns < 2; ++ns)
                    acc[ms][ns] = __builtin_amdgcn_wmma_f32_16x16x4_f32(
                        false, aFr[ms], false, bFr[ns], (short)0, acc[ms][ns],
                        false, false);
        }
        __syncthreads();
    }

    // epilogue: C layout — VGPR v: lanes 0-15 row=v, lanes 16-31 row=v+8
    #pragma unroll
    for (int ms = 0; ms < 2; ++ms)
        #pragma unroll
        for (int ns = 0; ns < 2; ++ns) {
            const int n  = n0 + wn * 32 + ns * 16 + l;
            const float bn = bias[n];
            #pragma unroll
            for (int v = 0; v < 8; ++v) {
                const int m = m0 + wm * 32 + ms * 16 + v + 8 * hi;
                xp[(size_t)m * G_ + n] = acc[ms][ns][v] + bn;
            }
        }
}

// --------------------- Kernel B: persistent recurrence ---------------------
// 8 workgroups; WG w owns h-columns [32w, 32w+32) => 96 gate columns (z/r/n).
// LDS: W_h slice 256x96 (stride 104, TDM preload), h 32x256 (stride 258, TDM
// per step), gates 32x96 (stride 100). Dynamic LDS = 152,320 B.
#define NB    8
#define HW    32
#define LWH   104
#define LHL   258
#define LGL   100
#define WH_F  0
#define HL_F  (256 * LWH)                  // 26624
#define GL_F  (256 * LWH + 32 * LHL)       // 34880
#define GB_LDS_FLOATS (256 * LWH + 32 * LHL + 32 * LGL)  // 38080 -> 152,320 B

__global__ __launch_bounds__(256)
void gru_recurrence(const float* __restrict__ xp,
                    const float* __restrict__ Wh_g,
                    float* __restrict__ hglob,
                    unsigned* __restrict__ counter,
                    float* __restrict__ out)
{
    extern __shared__ float smem[];
    float* Wh = smem + WH_F;
    float* hL = smem + HL_F;
    float* gL = smem + GL_F;

    const int tid  = threadIdx.x;
    const int wave = tid >> 5;
    const int lane = tid & 31;
    const int l    = lane & 15;
    const int hi   = lane >> 4;
    const int hc0  = blockIdx.x * HW;

    const int wave_s = __builtin_amdgcn_readfirstlane(wave);

    // one-time W_h slice preload via TDM: 3 gate tiles 32x256, stride G_,
    // LDS rows padded 32 -> 104 floats (pad_amount = 72 DWORDs -> code 71)
    if (wave_s == 0) {
        #pragma unroll
        for (int g = 0; g < 3; ++g)
            tdm_load_2d((WH_F + g * HW) * 4, Wh_g + g * H_ + hc0,
                        HW, H_, G_, 1, 4, 71);
    }

    // 24 jobs = 12 tiles (2M x 2N x 3 gates) x 2 K-halves; 3 per wave (ILP=3)
    int msub[3], nsub[3], gate[3], kb[3];
    #pragma unroll
    for (int i = 0; i < 3; ++i) {
        const int jw   = wave * 3 + i;
        const int tile = jw >> 1;
        kb[i]   = (jw & 1) * 128;
        msub[i] = tile & 1;
        nsub[i] = (tile >> 1) & 1;
        gate[i] = tile >> 2;
    }

    for (int t = 0; t < T_; ++t) {
        // TDM: pull h (32x256, 1D 8192 elems) into LDS, pad 256 -> 258 stride
        if (wave_s == 0)
            tdm_load_2d(HL_F * 4, hglob, 8192, 1, 8192, 1, 7, 1);
        // zero gate accumulator while the DMA flies
        for (int idx = tid; idx < 32 * LGL; idx += 256) gL[idx] = 0.0f;
        if (wave_s == 0)
            __builtin_amdgcn_s_wait_tensorcnt(0);  // h (and W_h on t=0) landed
        __syncthreads();

        // h @ W_h for this WG's 96 gate columns (fp32 WMMA, K split in 2)
        v8f acc[3] = {};
        #pragma unroll 4
        for (int kk = 0; kk < 32; ++kk) {
            #pragma unroll
            for (int i = 0; i < 3; ++i) {
                const int k = kb[i] + kk * 4;
                v2f aF = *(const v2f*)(hL + (msub[i] * 16 + l) * LHL + k + 2 * hi);
                v2f bF;
                const int col = gate[i] * HW + nsub[i] * 16 + l;
                bF.x = Wh[(k + 2 * hi)     * LWH + col];
                bF.y = Wh[(k + 2 * hi + 1) * LWH + col];
                acc[i] = __builtin_amdgcn_wmma_f32_16x16x4_f32(
                    false, aF, false, bF, (short)0, acc[i], false, false);
            }
        }
        // reduce K-partials via LDS float atomics (ds_add_f32)
        #pragma unroll
        for (int i = 0; i < 3; ++i) {
            const int col = gate[i] * HW + nsub[i] * 16 + l;
            #pragma unroll
            for (int v = 0; v < 8; ++v) {
                const int row = msub[i] * 16 + v + 8 * hi;
                atomicAdd(&gL[row * LGL + col], acc[i][v]);
            }
        }
        __syncthreads();

        // fused gate math + state update for our 32 h-columns (all 32 batches)
        #pragma unroll
        for (int i = 0; i < 4; ++i) {
            const int idx = tid + i * 256;     // 1024 elements
            const int b   = idx >> 5;
            const int c   = idx & 31;
            const int hc  = hc0 + c;
            const float hz = gL[b * LGL + c];
            const float hr = gL[b * LGL + HW + c];
            const float hn = gL[b * LGL + 2 * HW + c];
            const float* xpr = xp + ((size_t)b * T_ + t) * G_;
            const float xz = xpr[hc], xr = xpr[H_ + hc], xn = xpr[2 * H_ + hc];
            const float z = 1.0f / (1.0f + __expf(-(xz + hz)));
            const float r = 1.0f / (1.0f + __expf(-(xr + hr)));
            const float n = tanhf(xn + r * hn);
            const float hold = hL[b * LHL + hc];
            const float hnew = z * hold + (1.0f - z) * n;
            hglob[b * H_ + hc] = hnew;
            out[((size_t)b * T_ + t) * H_ + hc] = hnew;
        }

        // grid-wide barrier (monotonic counter, release/acquire at agent scope)
        __threadfence();
        __syncthreads();
        if (tid == 0) {
            __hip_atomic_fetch_add(counter, 1u, __ATOMIC_ACQ_REL,
                                   __HIP_MEMORY_SCOPE_AGENT);
            const unsigned target = (unsigned)NB * (unsigned)(t + 1);
            while (__hip_atomic_load(counter, __ATOMIC_ACQUIRE,
                                     __HIP_MEMORY_SCOPE_AGENT) < target)
                __builtin_amdgcn_s_sleep(2);
        }
        __syncthreads();
    }
}

// ------------------------------- launcher ----------------------------------
extern "C" void kernel_launch(void* const* d_in, const int* in_sizes, int n_in,
                              void* d_out, int out_size, void* d_ws, size_t ws_size,
                              hipStream_t stream) {
    const float* x    = (const float*)d_in[0];   // [B,T,D]
    const float* Wx   = (const float*)d_in[1];   // [D,3H]
    const float* Whp  = (const float*)d_in[2];   // [H,3H]
    const float* bias = (const float*)d_in[3];   // [3H]
    float* out = (float*)d_out;                  // [B,T,H] fp32

    char* ws = (char*)d_ws;
    float*    xpw     = (float*)ws;                       // 192 MB: x_proj
    const size_t xpB  = (size_t)M_ * G_ * sizeof(float);
    float*    hglob   = (float*)(ws + xpB);               // 32 KB: h state
    unsigned* counter = (unsigned*)(ws + xpB + (size_t)B_ * H_ * sizeof(float));

    // zero h0 and the barrier counter (graph-capture safe)
    hipMemsetAsync(hglob, 0, (size_t)B_ * H_ * sizeof(float) + 256, stream);

    // Kernel A: 1024 x 6 tiles of 64x128, 53,248 B dynamic LDS (double-buffered)
    const size_t shA = (size_t)GA_LDS_FLOATS * sizeof(float);
    hipFuncSetAttribute((const void*)xproj_gemm,
                        hipFuncAttributeMaxDynamicSharedMemorySize, (int)shA);
    dim3 gA(G_ / GA_NT, M_ / GA_MT);
    xproj_gemm<<<gA, 256, shA, stream>>>(x, Wx, bias, xpw);

    // Kernel B: 8 persistent WGs, 152,320 B dynamic LDS each
    const size_t shB = (size_t)GB_LDS_FLOATS * sizeof(float);
    hipFuncSetAttribute((const void*)gru_recurrence,
                        hipFuncAttributeMaxDynamicSharedMemorySize, (int)shB);
    gru_recurrence<<<NB, 256, shB, stream>>>(xpw, Whp, hglob, counter, out);
}